// BahdanauAttnDecoderRNN_3384434229615
// MI455X (gfx1250) — compile-verified
//
#include <hip/hip_runtime.h>
#include <hip/hip_bf16.h>
#include <math.h>

typedef __attribute__((ext_vector_type(2))) float v2f;
typedef __attribute__((ext_vector_type(8))) float v8f;

#define HID 1024
#define VOC 50257
#define LEN 2048

// ---------------------------------------------------------------------------
// WMMA f32 GEMV: out[m] = sum_k W[m,k] * x[k] (+ bias), W row-major [M,K].
// One wave per 16-row tile, V_WMMA_F32_16X16X4_F32 accumulate over K.
// A layout (16x4 f32): lanes 0-15 hold K=k,k+1 ; lanes 16-31 hold K=k+2,k+3.
// B is x broadcast across N -> every accumulator column equals the row dot.
// C/D layout: acc[j] lanes 0-15 = row m0+j ; lanes 16-31 = row m0+8+j.
// NT=true streams W with non-temporal hint (single-use data > L2 capacity),
// preserving L2 residency of the small reused weight matrices across replays.
// Requires K % 32 == 0 (true for all call sites: K = 1024 or 2048).
// ---------------------------------------------------------------------------
template <bool NT>
__global__ __launch_bounds__(256) void k_gemv_wmma(const float* __restrict__ W,
                                                   const float* __restrict__ x,
                                                   const float* __restrict__ bias,
                                                   float* __restrict__ out,
                                                   int M, int K) {
  const int lane = threadIdx.x & 31;
  const int wave = threadIdx.x >> 5;
  const int tile = blockIdx.x * 8 + wave;
  const int m0 = tile * 16;
  if (m0 >= M) return;                       // whole wave exits together
  int row = m0 + (lane & 15);
  if (row >= M) row = M - 1;                 // clamp for partial tail tile
  const int ko = (lane >> 4) << 1;           // 0 for lanes 0-15, 2 for 16-31
  const float* wr = W + (size_t)row * K + ko;
  const float* xr = x + ko;
  v8f acc = {0.f, 0.f, 0.f, 0.f, 0.f, 0.f, 0.f, 0.f};
  for (int k = 0; k < K; k += 32) {
    // ~1KB lookahead in this lane's row stream -> global_prefetch_b8
    __builtin_prefetch(wr + k + 256, 0, 0);
#pragma unroll
    for (int kk = 0; kk < 32; kk += 4) {
      v2f a = NT ? __builtin_nontemporal_load((const v2f*)(wr + k + kk))
                 : *(const v2f*)(wr + k + kk);
      v2f b = *(const v2f*)(xr + k + kk);
      acc = __builtin_amdgcn_wmma_f32_16x16x4_f32(false, a, false, b,
                                                  (short)0, acc, false, false);
    }
  }
  if ((lane & 15) == 0) {                    // lanes 0 and 16 write results
    const int base = m0 + ((lane >> 4) << 3);
#pragma unroll
    for (int j = 0; j < 8; ++j) {
      int r = base + j;
      if (r < M) out[r] = acc[j] + (bias ? bias[r] : 0.0f);
    }
  }
}

// ---- embedding lookup: xg[0:H) = emb[word, :] ------------------------------
__global__ void k_embed(const int* __restrict__ widx,
                        const float* __restrict__ emb,
                        float* __restrict__ xg) {
  int h = blockIdx.x * 256 + threadIdx.x;
  if (h < HID) xg[h] = emb[(size_t)widx[0] * HID + h];
}

// ---- u[k] = sum_h v[h] * attn_W[h, H+k]  (column matvec, coalesced in k) ---
__global__ void k_att_u(const float* __restrict__ attn_W,
                        const float* __restrict__ v,
                        float* __restrict__ u) {
  int k = blockIdx.x * 256 + threadIdx.x;
  if (k >= HID) return;
  float acc = 0.f;
  for (int h = 0; h < HID; ++h)
    acc += v[h] * attn_W[(size_t)h * (2 * HID) + HID + k];
  u[k] = acc;
}

// ---- softmax over L=2048 scores (single block, 256 thr x 8 elems) ----------
__global__ __launch_bounds__(256) void k_softmax(const float* __restrict__ sc,
                                                 float* __restrict__ w,
                                                 float* __restrict__ attn_out) {
  const int t = threadIdx.x;
  __shared__ float red[8];
  __shared__ float bmax, bsum;
  float vals[8];
  float m = -3.4e38f;
#pragma unroll
  for (int i = 0; i < 8; ++i) { vals[i] = sc[t + 256 * i]; m = fmaxf(m, vals[i]); }
  for (int off = 16; off > 0; off >>= 1) m = fmaxf(m, __shfl_xor(m, off));
  if ((t & 31) == 0) red[t >> 5] = m;
  __syncthreads();
  if (t == 0) { float mm = red[0]; for (int i = 1; i < 8; ++i) mm = fmaxf(mm, red[i]); bmax = mm; }
  __syncthreads();
  m = bmax;
  float s = 0.f;
#pragma unroll
  for (int i = 0; i < 8; ++i) { vals[i] = expf(vals[i] - m); s += vals[i]; }
  for (int off = 16; off > 0; off >>= 1) s += __shfl_xor(s, off);
  if ((t & 31) == 0) red[t >> 5] = s;
  __syncthreads();
  if (t == 0) { float sv = 0.f; for (int i = 0; i < 8; ++i) sv += red[i]; bsum = sv; }
  __syncthreads();
  const float inv = 1.0f / bsum;
#pragma unroll
  for (int i = 0; i < 8; ++i) {
    float wv = vals[i] * inv;
    w[t + 256 * i] = wv;
    attn_out[t + 256 * i] = wv;
  }
}

// ---- context partials: block b sums 64 rows of enc weighted by w -----------
__global__ __launch_bounds__(256) void k_ctx_part(const float* __restrict__ enc,
                                                  const float* __restrict__ w,
                                                  float* __restrict__ part) {
  const int b = blockIdx.x, t = threadIdx.x;
  float acc[4] = {0.f, 0.f, 0.f, 0.f};
  for (int l = b * 64; l < b * 64 + 64; ++l) {
    const float wl = w[l];
    const float* er = enc + (size_t)l * HID;
#pragma unroll
    for (int q = 0; q < 4; ++q) acc[q] += wl * er[t + 256 * q];
  }
#pragma unroll
  for (int q = 0; q < 4; ++q) part[(size_t)b * HID + t + 256 * q] = acc[q];
}

// ---- reduce 32 context partials; also drop context into xg[H:2H) -----------
__global__ void k_ctx_red(const float* __restrict__ part,
                          float* __restrict__ ctx,
                          float* __restrict__ xg_hi) {
  int h = blockIdx.x * 256 + threadIdx.x;
  if (h >= HID) return;
  float s = 0.f;
  for (int b = 0; b < 32; ++b) s += part[(size_t)b * HID + h];
  ctx[h] = s;
  xg_hi[h] = s;
}

// ---- GRU gates + assemble x2 = [h_new ; context] ---------------------------
__global__ void k_gru(const float* __restrict__ gi, const float* __restrict__ gh,
                      const float* __restrict__ hprev, const float* __restrict__ ctx,
                      float* __restrict__ hnew_out, float* __restrict__ x2) {
  int h = blockIdx.x * 256 + threadIdx.x;
  if (h >= HID) return;
  float r = 1.f / (1.f + expf(-(gi[h] + gh[h])));
  float z = 1.f / (1.f + expf(-(gi[HID + h] + gh[HID + h])));
  float n = tanhf(gi[2 * HID + h] + r * gh[2 * HID + h]);
  float hp = hprev[h];
  float hn = (1.f - z) * n + z * hp;
  hnew_out[h] = hn;
  x2[h] = hn;
  x2[HID + h] = ctx[h];
}

// ---- log-sum-exp over logits: online (max,sum) partials, fixed tree --------
__device__ __forceinline__ void lse_comb(float& m, float& s, float m2, float s2) {
  if (m2 > m) { s = s2 + s * expf(m - m2); m = m2; }
  else        { s = s + s2 * expf(m2 - m); }
}

__global__ __launch_bounds__(256) void k_lse_part(const float* __restrict__ logits,
                                                  float* __restrict__ part) {
  const int g = blockIdx.x * 256 + threadIdx.x;   // 0..16383, always < VOC
  float m = logits[g];
  float s = 1.0f;
  for (int i = g + 16384; i < VOC; i += 16384) {
    float xv = logits[i];
    if (xv > m) { s = s * expf(m - xv) + 1.0f; m = xv; }
    else        { s += expf(xv - m); }
  }
  for (int off = 16; off > 0; off >>= 1) {
    float m2 = __shfl_xor(m, off);
    float s2 = __shfl_xor(s, off);
    lse_comb(m, s, m2, s2);
  }
  __shared__ float sm[8], ss[8];
  const int t = threadIdx.x;
  if ((t & 31) == 0) { sm[t >> 5] = m; ss[t >> 5] = s; }
  __syncthreads();
  if (t == 0) {
    for (int i = 1; i < 8; ++i) lse_comb(m, s, sm[i], ss[i]);
    part[2 * blockIdx.x] = m;
    part[2 * blockIdx.x + 1] = s;
  }
}

__global__ void k_lse_fin(const float* __restrict__ part, float* __restrict__ stats) {
  if (threadIdx.x == 0) {
    float m = part[0], s = part[1];
    for (int i = 1; i < 64; ++i) lse_comb(m, s, part[2 * i], part[2 * i + 1]);
    stats[0] = m;
    stats[1] = logf(s);
  }
}

__global__ void k_apply(float* __restrict__ out, const float* __restrict__ stats) {
  int i = blockIdx.x * 256 + threadIdx.x;
  if (i < VOC) out[i] -= (stats[0] + stats[1]);
}

// ---------------------------------------------------------------------------
extern "C" void kernel_launch(void* const* d_in, const int* in_sizes, int n_in,
                              void* d_out, int out_size, void* d_ws, size_t ws_size,
                              hipStream_t stream) {
  const int*   word_input  = (const int*)d_in[0];
  const float* last_hidden = (const float*)d_in[1];
  const float* enc         = (const float*)d_in[2];
  const float* emb         = (const float*)d_in[3];
  const float* attn_W      = (const float*)d_in[4];
  /* d_in[5] = attn_b: constant shift under softmax -> mathematically unused */
  const float* v           = (const float*)d_in[6];
  const float* W_ih        = (const float*)d_in[7];
  const float* W_hh        = (const float*)d_in[8];
  const float* b_ih        = (const float*)d_in[9];
  const float* b_hh        = (const float*)d_in[10];
  const float* out_W       = (const float*)d_in[11];
  const float* out_b       = (const float*)d_in[12];
  float* out = (float*)d_out;            // [log_probs(V) | h_new(H) | attn(L)]
  float* ws  = (float*)d_ws;

  float* xg    = ws;                     // 2048 : [embedded ; context]
  float* u     = ws + 2048;              // 1024 : attn_W[:,H:]^T v
  float* sc    = ws + 3072;              // 2048 : raw scores
  float* w     = ws + 5120;              // 2048 : attention weights
  float* ctx   = ws + 7168;              // 1024 : context vector
  float* cpart = ws + 8192;              // 32*1024 context partials
  float* gi    = ws + 40960;             // 3072
  float* gh    = ws + 44032;             // 3072
  float* x2    = ws + 47104;             // 2048 : [h_new ; context]
  float* lpart = ws + 49152;             // 128  : 64 (max,sum) pairs
  float* stats = ws + 49280;             // 2    : {max, log(sum)}

  auto gemv_blocks = [](int M) { int tiles = (M + 15) / 16; return (tiles + 7) / 8; };

  k_embed<<<4, 256, 0, stream>>>(word_input, emb, xg);
  k_att_u<<<4, 256, 0, stream>>>(attn_W, v, u);
  // scores = enc[2048x1024] @ u                                  (WMMA, L2-resident)
  k_gemv_wmma<false><<<gemv_blocks(LEN), 256, 0, stream>>>(enc, u, nullptr, sc, LEN, HID);
  k_softmax<<<1, 256, 0, stream>>>(sc, w, out + VOC + HID);
  k_ctx_part<<<32, 256, 0, stream>>>(enc, w, cpart);
  k_ctx_red<<<4, 256, 0, stream>>>(cpart, ctx, xg + HID);
  // gi = W_ih[3072x2048] @ [we;ctx] + b_ih                       (WMMA, L2-resident)
  k_gemv_wmma<false><<<gemv_blocks(3 * HID), 256, 0, stream>>>(W_ih, xg, b_ih, gi, 3 * HID, 2 * HID);
  // gh = W_hh[3072x1024] @ h + b_hh                              (WMMA, L2-resident)
  k_gemv_wmma<false><<<gemv_blocks(3 * HID), 256, 0, stream>>>(W_hh, last_hidden, b_hh, gh, 3 * HID, HID);
  k_gru<<<4, 256, 0, stream>>>(gi, gh, last_hidden, ctx, out + VOC, x2);
  // logits = out_W[50257x2048] @ [h_new;ctx] + out_b
  //   412 MB single-use stream > L2 -> non-temporal              (WMMA, NT)
  k_gemv_wmma<true><<<gemv_blocks(VOC), 256, 0, stream>>>(out_W, x2, out_b, out, VOC, 2 * HID);
  k_lse_part<<<64, 256, 0, stream>>>(out, lpart);
  k_lse_fin<<<1, 64, 0, stream>>>(lpart, stats);
  k_apply<<<(VOC + 255) / 256, 256, 0, stream>>>(out, stats);
}